// myCBOWNS_35716948034467
// MI455X (gfx1250) — compile-verified
//
#include <hip/hip_runtime.h>

typedef __attribute__((ext_vector_type(2))) float v2f;
typedef __attribute__((ext_vector_type(8))) float v8f;

// Problem constants (match reference)
#define VOCAB 200000
#define DIM   50
#define BTOT  16384
#define CTX   10
#define NNEG  10

#define TILE   16          // batch rows per wave
#define KPAD   52          // 50 dims padded to 13 chunks of K=4
#define NCHUNK 13
#define NPAIR  (TILE * (DIM / 2))      // 400 float2 elements per emb tile
#define NITER  ((NPAIR + 31) / 32)     // 13 async b64 issues per gather
#define WPB    4           // waves per block (128 threads, wave32)
#define NBLK   (BTOT / (TILE * WPB))   // 256 blocks -> 1024 waves == 1024 tiles

// Issue one embedding-tile gather (16 rows x 50 f32) as 13 async global->LDS
// b64 loads. Lanes whose flattened element index exceeds 399 clamp to row 0 and
// produce benign duplicate writes, so EXEC stays full and ASYNCcnt bumps by
// exactly NITER per call.
__device__ __forceinline__ void issue_gather(
    const float* __restrict__ o_emb, const int* __restrict__ tgt,
    const int* __restrict__ neg, int j, int b0, float* embBuf, int lane)
{
    const float* gaddr[NITER];
    unsigned int laddr[NITER];
    #pragma unroll
    for (int i = 0; i < NITER; ++i) {
        int e2 = i * 32 + lane;           // float2 element index
        int r  = e2 / 25;                 // row in tile (25 pairs per row)
        int p  = e2 - r * 25;             // pair within row
        if (r >= TILE) r = 0;             // clamp -> duplicate row-0 write
        const int bb  = b0 + r;
        const int wid = (j == 0) ? tgt[bb] : neg[bb * NNEG + (j - 1)];
        gaddr[i] = o_emb + (size_t)wid * DIM + 2 * p;
        laddr[i] = (unsigned int)(size_t)(embBuf + r * KPAD + 2 * p);
    }
    #pragma unroll
    for (int i = 0; i < NITER; ++i)
        asm volatile("global_load_async_to_lds_b64 %0, %1, off"
                     :: "v"(laddr[i]), "v"(gaddr[i]) : "memory");
}

__global__ __launch_bounds__(128) void cbow_ns_wmma_kernel(
    const float* __restrict__ i_emb,   // [VOCAB+1, DIM]
    const float* __restrict__ o_emb,   // [VOCAB+1, DIM]
    const int*   __restrict__ tgt,     // [B]
    const int*   __restrict__ ctx,     // [B, CTX]
    const int*   __restrict__ neg,     // [B, NNEG]
    float*       __restrict__ partial) // [1024] per-wave partial losses
{
    // [wave][0]=avg tile, [wave][1..2]=double-buffered emb tile  (40 KB/block)
    __shared__ __align__(16) float smem[WPB][3][TILE * KPAD];

    const int lane = threadIdx.x & 31;
    const int wav  = threadIdx.x >> 5;
    const int tile = blockIdx.x * WPB + wav;   // exactly 1024 tiles
    const int b0   = tile * TILE;

    float* avgT = smem[wav][0];
    float* emb0 = smem[wav][1];
    float* emb1 = smem[wav][2];

    const int h = lane >> 4;   // selects K pair {0,1} vs {2,3}
    const int m = lane & 15;   // matrix row (== batch row within tile)

    // zero the K-padding slots (d=50,51) of both emb buffers once
    if (lane < TILE) {
        emb0[lane * KPAD + 50] = 0.f; emb0[lane * KPAD + 51] = 0.f;
        emb1[lane * KPAD + 50] = 0.f; emb1[lane * KPAD + 51] = 0.f;
    }

    // ---- Phase A: average of 10 context embeddings -> LDS tile [16][52]
    for (int r = 0; r < TILE; ++r) {
        float a0 = 0.f, a1 = 0.f;
        const int cb = (b0 + r) * CTX;
        #pragma unroll
        for (int c = 0; c < CTX; ++c) {
            const int wid = ctx[cb + c];
            const float* row = i_emb + (size_t)wid * DIM;
            a0 += row[lane];                            // dims 0..31
            if (lane < DIM - 32) a1 += row[lane + 32];  // dims 32..49
        }
        avgT[r * KPAD + lane] = a0 * 0.1f;
        float v1 = (lane < DIM - 32) ? a1 * 0.1f : 0.f; // zero-pads dims 50,51
        if (lane < KPAD - 32) avgT[r * KPAD + 32 + lane] = v1;
    }
    asm volatile("s_wait_dscnt 0" ::: "memory");

    float lossAcc = 0.f;

    // ---- Phase B: double-buffered async gather overlapped with WMMA
    issue_gather(o_emb, tgt, neg, /*j=*/0, b0, emb0, lane);

    for (int j = 0; j <= NNEG; ++j) {        // j==0 -> target, j>=1 -> negatives
        if (j < NNEG)
            issue_gather(o_emb, tgt, neg, j + 1, b0, ((j + 1) & 1) ? emb1 : emb0, lane);

        // async loads complete in order: <=NITER outstanding => current buffer ready
        if (j < NNEG)
            asm volatile("s_wait_asynccnt 13" ::: "memory");
        else
            asm volatile("s_wait_asynccnt 0" ::: "memory");

        const float* embT = (j & 1) ? emb1 : emb0;

        // 16x16 = avg(16x52) x emb^T(52x16) via 13 chained F32 WMMAs
        v8f acc = {};
        #pragma unroll
        for (int c = 0; c < NCHUNK; ++c) {
            const int off = 4 * c + 2 * h;
            v2f a = *(const v2f*)(avgT + m * KPAD + off);
            v2f b = *(const v2f*)(embT + m * KPAD + off);
            acc = __builtin_amdgcn_wmma_f32_16x16x4_f32(
                      false, a, false, b, (short)0, acc, false, false);
        }

        // diagonal: row m<8 -> lane m, acc[m]; row m>=8 -> lane m+16, acc[m-8]
        const bool valid = (lane < 8) || (lane >= 24);
        const int  idx   = (lane < 8) ? lane : (lane - 24);
        float s = acc[0];
        #pragma unroll
        for (int k = 1; k < 8; ++k) s = (idx == k) ? acc[k] : s;
        if (j != 0) s = -s;                     // negatives use -dot

        // -log_sigmoid(s) == softplus(-s), numerically stable
        const float x = -s;
        const float contrib = fmaxf(x, 0.f) + __logf(1.f + __expf(-fabsf(x)));
        if (valid) lossAcc += contrib;
    }

    // wave32 butterfly reduction, lane 0 writes the wave's partial
    #pragma unroll
    for (int off = 16; off > 0; off >>= 1)
        lossAcc += __shfl_xor(lossAcc, off, 32);
    if (lane == 0) partial[tile] = lossAcc;
}

// Deterministic fixed-order reduction of 1024 partials -> scalar loss
__global__ __launch_bounds__(256) void cbow_ns_reduce_kernel(
    const float* __restrict__ partial, float* __restrict__ out)
{
    __shared__ float sm[256];
    const int t = threadIdx.x;
    float v = partial[t] + partial[t + 256] + partial[t + 512] + partial[t + 768];
    sm[t] = v;
    __syncthreads();
    for (int s = 128; s > 0; s >>= 1) {
        if (t < s) sm[t] += sm[t + s];
        __syncthreads();
    }
    if (t == 0) out[0] = sm[0];
}

extern "C" void kernel_launch(void* const* d_in, const int* in_sizes, int n_in,
                              void* d_out, int out_size, void* d_ws, size_t ws_size,
                              hipStream_t stream) {
    const float* i_emb = (const float*)d_in[0];
    const float* o_emb = (const float*)d_in[1];
    const int*   tgt   = (const int*)d_in[2];
    const int*   ctxp  = (const int*)d_in[3];
    const int*   negp  = (const int*)d_in[4];
    float* partial = (float*)d_ws;             // 1024 floats of scratch
    float* out     = (float*)d_out;

    cbow_ns_wmma_kernel<<<NBLK, 128, 0, stream>>>(i_emb, o_emb, tgt, ctxp, negp, partial);
    cbow_ns_reduce_kernel<<<1, 256, 0, stream>>>(partial, out);
}